// TTTConv_13099650253054
// MI455X (gfx1250) — compile-verified
//
#include <hip/hip_runtime.h>
#include <cstdint>
#include <cstddef>

// Problem constants (match reference)
#define B_ 4
#define N_ 4096
#define D_ 2048
#define K_ 4

// Tiling
#define THREADS    256
#define VEC        4                    // floats per thread per row (b128)
#define DCHUNK     (THREADS * VEC)      // 1024 channels per block
#define CH         8                    // rows per async chunk
#define LN         64                   // rows (n) per block
#define NCHUNKS    (LN / CH)
#define ROWBYTES   (THREADS * 16)       // 4096 B per staged row
#define CHUNKBYTES (CH * ROWBYTES)      // 32768 B per chunk buffer

typedef float v4f __attribute__((ext_vector_type(4)));

__global__ __launch_bounds__(THREADS)
void TTTConv_main_kernel(const float* __restrict__ x,
                         const float* __restrict__ w,
                         const float* __restrict__ bias,
                         float* __restrict__ out)
{
    // double-buffered staging: 2 * 8 rows * 4 KB = 64 KB LDS
    __shared__ float lds_buf[2 * CH * THREADS * VEC];

    const int tid = threadIdx.x;
    unsigned bx  = blockIdx.x;
    const int dblk   = bx & 1;  bx >>= 1;          // D_/DCHUNK == 2
    const int b      = bx & (B_ - 1);              // B_ == 4
    const int nchunk = bx >> 2;
    const int n0 = nchunk * LN;
    const int d0 = dblk * DCHUNK + tid * VEC;

    // per-channel taps: conv_weight is (D,K) row-major -> 4 contiguous floats per channel
    const v4f wc0 = *(const v4f*)(w + (size_t)(d0 + 0) * K_);
    const v4f wc1 = *(const v4f*)(w + (size_t)(d0 + 1) * K_);
    const v4f wc2 = *(const v4f*)(w + (size_t)(d0 + 2) * K_);
    const v4f wc3 = *(const v4f*)(w + (size_t)(d0 + 3) * K_);
    const v4f bi  = *(const v4f*)(bias + d0);

    // sliding window x[n0-3 .. n0-1] (zero-padded at sequence start)
    v4f xm3 = (v4f)0.f, xm2 = (v4f)0.f, xm1 = (v4f)0.f;
    const size_t xbase = (size_t)b * N_ * D_;
    if (n0 > 0) {
        xm3 = *(const v4f*)(x + xbase + (size_t)(n0 - 3) * D_ + d0);
        xm2 = *(const v4f*)(x + xbase + (size_t)(n0 - 2) * D_ + d0);
        xm1 = *(const v4f*)(x + xbase + (size_t)(n0 - 1) * D_ + d0);
    }

    // ---- async global->LDS pipeline (CDNA5 ASYNCcnt path) ----
    const uint32_t lds_base = (uint32_t)(uintptr_t)(&lds_buf[0]);
    const uint64_t sbase    = (uint64_t)(uintptr_t)x;
    const uint32_t voff0    = (uint32_t)((xbase + (size_t)n0 * D_ + d0) * sizeof(float));

    auto issue_chunk = [&](int c) {
        const uint32_t lds_off = lds_base + (uint32_t)((c & 1) * CHUNKBYTES) + (uint32_t)tid * 16u;
        uint32_t go = voff0 + (uint32_t)(c * CH) * (uint32_t)(D_ * sizeof(float));
        #pragma unroll
        for (int r = 0; r < CH; ++r) {
            // copy from address saddr + vaddr into LDS at vdst  (16 bytes/lane, tracked by ASYNCcnt)
            asm volatile("global_load_async_to_lds_b128 %0, %1, %2"
                         :
                         : "v"(lds_off + (uint32_t)(r * ROWBYTES)), "v"(go), "s"(sbase)
                         : "memory");
            go += (uint32_t)(D_ * sizeof(float));
        }
    };

    issue_chunk(0);

    float* outp = out + xbase + (size_t)n0 * D_ + d0;
    for (int c = 0; c < NCHUNKS; ++c) {
        if (c + 1 < NCHUNKS) {
            issue_chunk(c + 1);                                  // prefetch next chunk
            asm volatile("s_wait_asynccnt 0x8" ::: "memory");    // oldest 8 (chunk c) done
        } else {
            asm volatile("s_wait_asynccnt 0x0" ::: "memory");    // drain
        }

        const float* src = &lds_buf[(c & 1) * (CH * THREADS * VEC) + tid * VEC];
        #pragma unroll
        for (int r = 0; r < CH; ++r) {
            const v4f xc = *(const v4f*)(src + r * (THREADS * VEC));
            v4f y;
            y.x = fmaf(wc0.w, xc.x, fmaf(wc0.z, xm1.x, fmaf(wc0.y, xm2.x, fmaf(wc0.x, xm3.x, bi.x))));
            y.y = fmaf(wc1.w, xc.y, fmaf(wc1.z, xm1.y, fmaf(wc1.y, xm2.y, fmaf(wc1.x, xm3.y, bi.y))));
            y.z = fmaf(wc2.w, xc.z, fmaf(wc2.z, xm1.z, fmaf(wc2.y, xm2.z, fmaf(wc2.x, xm3.z, bi.z))));
            y.w = fmaf(wc3.w, xc.w, fmaf(wc3.z, xm1.w, fmaf(wc3.y, xm2.w, fmaf(wc3.x, xm3.w, bi.w))));
            xm3 = xm2; xm2 = xm1; xm1 = xc;
            __builtin_nontemporal_store(y, (v4f*)outp);          // streamed output, keep L2 for x
            outp += D_;
        }
    }
}

// conv_states[b,d,k] = x[b, N-4+k, d]   (B*D*K = 32K floats)
__global__ __launch_bounds__(256)
void TTTConv_states_kernel(const float* __restrict__ x, float* __restrict__ out2)
{
    const int t = blockIdx.x * blockDim.x + threadIdx.x;  // over B_*D_
    if (t >= B_ * D_) return;
    const int b = t / D_;
    const int d = t - b * D_;
    const float* xp = x + (size_t)b * N_ * D_ + (size_t)(N_ - K_) * D_ + d;
    v4f s;
    s.x = xp[0 * D_];
    s.y = xp[1 * D_];
    s.z = xp[2 * D_];
    s.w = xp[3 * D_];
    *(v4f*)(out2 + (size_t)t * K_) = s;
}

extern "C" void kernel_launch(void* const* d_in, const int* in_sizes, int n_in,
                              void* d_out, int out_size, void* d_ws, size_t ws_size,
                              hipStream_t stream)
{
    (void)in_sizes; (void)n_in; (void)out_size; (void)d_ws; (void)ws_size;
    const float* x    = (const float*)d_in[0];
    const float* w    = (const float*)d_in[1];
    const float* bias = (const float*)d_in[2];
    float* out        = (float*)d_out;

    const int grid_main = (D_ / DCHUNK) * B_ * (N_ / LN);     // 2 * 4 * 64 = 512 blocks
    TTTConv_main_kernel<<<grid_main, THREADS, 0, stream>>>(x, w, bias, out);

    float* out_states = out + (size_t)B_ * N_ * D_;
    TTTConv_states_kernel<<<(B_ * D_ + 255) / 256, 256, 0, stream>>>(x, out_states);
}